// ConstrainedDeformableConv2d4DoF_61151744360471
// MI455X (gfx1250) — compile-verified
//
#include <hip/hip_runtime.h>

typedef __attribute__((ext_vector_type(2))) float v2f;
typedef __attribute__((ext_vector_type(8))) float v8f;

#define HW   128
#define CIN  64
#define COUT 64
#define KDIM 576   // CIN * 9
#define PLANE (HW*HW)

// ---------------------------------------------------------------------------
// Kernel 1: offset conv  o[b,j,h,w] = conv(x, offset_w)[j] + offset_b[j], j=0..3
// One thread per (b,h,w); computes all 4 offset channels to reuse x loads.
// ---------------------------------------------------------------------------
__global__ void __launch_bounds__(256)
offset_conv_kernel(const float* __restrict__ x,
                   const float* __restrict__ ow,
                   const float* __restrict__ ob,
                   float* __restrict__ o) {
  int idx = blockIdx.x * blockDim.x + threadIdx.x;     // 8*128*128 = 131072
  int w = idx & 127, h = (idx >> 7) & 127, b = idx >> 14;
  float a0 = ob[0], a1 = ob[1], a2 = ob[2], a3 = ob[3];
  const float* xb = x + (size_t)b * CIN * PLANE;
  for (int c = 0; c < CIN; ++c) {
    const float* xp = xb + c * PLANE;
    #pragma unroll
    for (int ki = 0; ki < 3; ++ki) {
      int y = h - 1 + ki;
      if (y < 0 || y >= HW) continue;
      #pragma unroll
      for (int kj = 0; kj < 3; ++kj) {
        int xx = w - 1 + kj;
        if (xx < 0 || xx >= HW) continue;
        float v = xp[y * HW + xx];
        int wi = c * 9 + ki * 3 + kj;
        a0 = fmaf(v, ow[0 * KDIM + wi], a0);
        a1 = fmaf(v, ow[1 * KDIM + wi], a1);
        a2 = fmaf(v, ow[2 * KDIM + wi], a2);
        a3 = fmaf(v, ow[3 * KDIM + wi], a3);
      }
    }
  }
  int base = ((b * 4) * HW + h) * HW + w;
  o[base]             = a0;
  o[base + PLANE]     = a1;
  o[base + 2 * PLANE] = a2;
  o[base + 3 * PLANE] = a3;
}

// ---------------------------------------------------------------------------
// Kernel 2: transpose weight (COUT, KDIM) -> Wt (KDIM, COUT) so that WMMA
// A-fragment loads are lane-contiguous (64B segments).
// ---------------------------------------------------------------------------
__global__ void __launch_bounds__(256)
transpose_w_kernel(const float* __restrict__ wsrc, float* __restrict__ wt) {
  int i = blockIdx.x * blockDim.x + threadIdx.x;
  if (i >= COUT * KDIM) return;
  int o = i / KDIM, kk = i - o * KDIM;
  wt[kk * COUT + o] = wsrc[i];
}

__device__ __forceinline__ float sample_at(const float* __restrict__ xp, int y, int xx) {
  if (y < 0 || y >= HW || xx < 0 || xx >= HW) return 0.0f;
  return xp[y * HW + xx];
}

// ---------------------------------------------------------------------------
// Kernel 3: fused bilinear sampling + GEMM (M=64, K=576, N=16 per block).
// 128 threads = 4 wave32; wave wv computes out-channels [16*wv, 16*wv+16).
// ---------------------------------------------------------------------------
__global__ void __launch_bounds__(128)
deform_gemm_kernel(const float* __restrict__ x,
                   const float* __restrict__ o,
                   const float* __restrict__ wt,
                   float* __restrict__ out) {
  __shared__ float sB[KDIM * 16];     // 36 KB: B-tile (K-major, 16 pixels wide)
  __shared__ float sLy[144], sLx[144];
  __shared__ int   sY0[144], sX0[144];

  int tid = threadIdx.x;
  int blk = blockIdx.x;               // 8 wtiles * 128 rows * 8 batches = 8192
  int w0  = (blk & 7) * 16;
  int h   = (blk >> 3) & 127;
  int b   = blk >> 10;

  // tap -> offset-channel selector, 3 bits/tap, 7 == zero
  // dy: {t,b,z,t,b,l,z,z,r}   dx: {z,t,b,z,l,l,z,r,r}
  const unsigned DYC = (0u<<0)|(1u<<3)|(7u<<6)|(0u<<9)|(1u<<12)|(2u<<15)|(7u<<18)|(7u<<21)|(3u<<24);
  const unsigned DXC = (7u<<0)|(0u<<3)|(1u<<6)|(7u<<9)|(2u<<12)|(2u<<15)|(7u<<18)|(3u<<21)|(3u<<24);

  // ---- Phase 1: bilinear coords per (pixel, tap), shared across channels ----
  const float* obase = o + (size_t)b * 4 * PLANE;
  for (int pk = tid; pk < 144; pk += 128) {
    int p = pk & 15, tap = pk >> 4;
    int w = w0 + p;
    int pix = h * HW + w;
    float ot = obase[pix];
    float obm = obase[pix + PLANE];
    float ol = obase[pix + 2 * PLANE];
    float orr = obase[pix + 3 * PLANE];
    int sy = (DYC >> (3 * tap)) & 7;
    int sx = (DXC >> (3 * tap)) & 7;
    float dy = (sy == 0) ? ot : (sy == 1) ? obm : (sy == 2) ? ol : (sy == 3) ? orr : 0.0f;
    float dx = (sx == 0) ? ot : (sx == 1) ? obm : (sx == 2) ? ol : (sx == 3) ? orr : 0.0f;
    int ki = tap / 3, kj = tap - ki * 3;
    float py = (float)(h - 1 + ki) + dy;
    float px = (float)(w - 1 + kj) + dx;
    float y0f = floorf(py), x0f = floorf(px);
    sLy[pk] = py - y0f;
    sLx[pk] = px - x0f;
    sY0[pk] = (int)y0f;
    sX0[pk] = (int)x0f;
  }
  __syncthreads();

  // ---- Phase 2: gather + blend 576 x 16 fp32 B-tile into LDS ----
  const float* xb = x + (size_t)b * CIN * PLANE;
  for (int c = 0; c < CIN; ++c) {
    const float* xp = xb + c * PLANE;
    for (int pk = tid; pk < 144; pk += 128) {
      int y0 = sY0[pk], x0 = sX0[pk];
      float ly = sLy[pk], lx = sLx[pk];
      float v00 = sample_at(xp, y0,     x0    );
      float v01 = sample_at(xp, y0,     x0 + 1);
      float v10 = sample_at(xp, y0 + 1, x0    );
      float v11 = sample_at(xp, y0 + 1, x0 + 1);
      float val = v00 * (1.0f - ly) * (1.0f - lx)
                + v01 * (1.0f - ly) * lx
                + v10 * ly * (1.0f - lx)
                + v11 * ly * lx;
      int p = pk & 15, tap = pk >> 4;
      sB[(c * 9 + tap) * 16 + p] = val;
    }
  }
  __syncthreads();

  // ---- Phase 3: fp32 WMMA GEMM, K = 576 in steps of 4 ----
  int wv   = tid >> 5;        // wave id 0..3 -> out-channel tile
  int lane = tid & 31;
  int mrow = lane & 15;       // A: M row / B: N column
  int hi   = lane >> 4;       // half-wave selects K pair {0,1} vs {2,3}
  int o0   = wv * 16;

  v8f acc = {};
  for (int kk = 0; kk < KDIM; kk += 4) {
    int ka = kk + 2 * hi;
    // A (16x4): lane m = mrow, vgpr0/1 = cols ka, ka+1 (ISA 32-bit A layout)
    v2f a = { wt[ka * COUT + o0 + mrow], wt[(ka + 1) * COUT + o0 + mrow] };
    // B (4x16): lane n = mrow, vgpr0/1 = rows ka, ka+1 (mirrored layout)
    v2f bb = { sB[ka * 16 + mrow], sB[(ka + 1) * 16 + mrow] };
    acc = __builtin_amdgcn_wmma_f32_16x16x4_f32(
        /*neg_a=*/false, a, /*neg_b=*/false, bb,
        /*c_mod=*/(short)0, acc, /*reuse_a=*/false, /*reuse_b=*/false);
  }

  // ---- Store: C layout VGPR r -> M = r + 8*hi, N = mrow ----
  float* op = out + (((size_t)(b * COUT + o0 + 8 * hi)) * HW + h) * HW + w0 + mrow;
  #pragma unroll
  for (int r = 0; r < 8; ++r) {
    op[(size_t)r * PLANE] = acc[r];
  }
}

// ---------------------------------------------------------------------------
extern "C" void kernel_launch(void* const* d_in, const int* in_sizes, int n_in,
                              void* d_out, int out_size, void* d_ws, size_t ws_size,
                              hipStream_t stream) {
  const float* x   = (const float*)d_in[0];   // (8,64,128,128)
  const float* ow  = (const float*)d_in[1];   // (4,64,3,3)
  const float* ob  = (const float*)d_in[2];   // (4,)
  const float* wgt = (const float*)d_in[3];   // (64,64,3,3)
  float* out = (float*)d_out;                 // (8,64,128,128)

  float* o_ws = (float*)d_ws;                 // 524288 floats (offset maps)
  float* wt   = o_ws + 8 * 4 * PLANE;         // 36864 floats (transposed weight)

  offset_conv_kernel<<<512, 256, 0, stream>>>(x, ow, ob, o_ws);
  transpose_w_kernel<<<(COUT * KDIM + 255) / 256, 256, 0, stream>>>(wgt, wt);
  deform_gemm_kernel<<<8192, 128, 0, stream>>>(x, o_ws, wt, out);
}